// DynaConv_64115271794993
// MI455X (gfx1250) — compile-verified
//
#include <hip/hip_runtime.h>

typedef float v2f __attribute__((ext_vector_type(2)));
typedef float v8f __attribute__((ext_vector_type(8)));

#define BATCH 16
#define LATD  512
#define FIN   128
#define FOUT  128
#define HH    64
#define WW    64
#define KSZ   (FOUT * FIN * 9)   // 147456
#define NCOL  (KSZ + FOUT)       // 147584
#define KCHUNK 32                // K-step staged per wave in stage 1
#define CF     16                // fin chunk in stage 2

// ---- CDNA5 async LDS copy helpers (ASYNCcnt-tracked, no VGPR round-trip) ----
__device__ __forceinline__ void async_copy_b128(unsigned lds_addr, const float* gaddr) {
    asm volatile("global_load_async_to_lds_b128 %0, %1, off"
                 :: "v"(lds_addr), "v"(gaddr) : "memory");
}
__device__ __forceinline__ void async_copy_b32(unsigned lds_addr, const float* gaddr) {
    asm volatile("global_load_async_to_lds_b32 %0, %1, off"
                 :: "v"(lds_addr), "v"(gaddr) : "memory");
}
__device__ __forceinline__ void wait_async0() {
    asm volatile("s_wait_asynccnt 0x0" ::: "memory");
}
// Generic->LDS byte address: low 32 bits of the flat LDS-aperture address are
// the LDS byte offset (ISA 10.2).
__device__ __forceinline__ unsigned lds_addr_of(const void* p) {
    return (unsigned)(unsigned long long)p;
}

// ---------------------------------------------------------------------------
// Stage 1: ks[m][n] = sum_k lat[m][k] * W[n][k] + bias[n]
// HBM-bound (streams 302 MB of W). 8 waves/block, each wave owns a 16-column
// tile of ks. lat (16x512 = 32 KB) staged once in LDS; W streamed via
// double-buffered async b128 copies into per-wave LDS slabs. fp32 WMMA 16x16x4.
// ---------------------------------------------------------------------------
__global__ __launch_bounds__(256)
void hyper_gemm_kernel(const float* __restrict__ lat,
                       const float* __restrict__ Wm,
                       const float* __restrict__ bias,
                       float* __restrict__ ks)
{
    __shared__ float lds_lat[BATCH * LATD];         // 32 KB, shared by all waves
    __shared__ float lds_w[8][2][16 * KCHUNK];      // 2 x 2 KB per wave (double buffer)

    const int tid  = threadIdx.x;
    const int wave = tid >> 5;
    const int lane = tid & 31;
    const int lg   = lane >> 4;   // lane group: K base offset 2*lg
    const int ll   = lane & 15;

    // Cooperative, fully coalesced staging of lat (8192 floats / 256 thr).
    for (int i = tid; i < BATCH * LATD; i += 256)
        lds_lat[i] = lat[i];
    __syncthreads();

    const int ntile = blockIdx.x * 8 + wave;        // 0 .. 9223
    const int c0    = ntile * 16;                   // first output column

    // C init = bias[n] broadcast over the batch (M) dimension.
    v8f acc;
    {
        const float bv = bias[c0 + ll];
#pragma unroll
        for (int r = 0; r < 8; ++r) acc[r] = bv;
    }

    const float* wbase = Wm + (size_t)c0 * LATD;

    // Per-lane source/dest pattern for one 16x32 slab (512 floats, 4 x b128/lane).
    const int e0   = lane * 4;                      // element within 128-float stripe
    const int row0 = e0 >> 5;                       // 0..3 (stripe i adds 4 rows)
    const int kk0  = e0 & (KCHUNK - 1);

    // Prime buffer 0.
#pragma unroll
    for (int i = 0; i < 4; ++i)
        async_copy_b128(lds_addr_of(&lds_w[wave][0][i * 128 + e0]),
                        wbase + (size_t)(row0 + i * 4) * LATD + kk0);

    int buf = 0;
    for (int kc = 0; kc < LATD; kc += KCHUNK) {
        wait_async0();                               // current slab resident
        if (kc + KCHUNK < LATD) {                    // issue next slab async
#pragma unroll
            for (int i = 0; i < 4; ++i)
                async_copy_b128(lds_addr_of(&lds_w[wave][buf ^ 1][i * 128 + e0]),
                                wbase + (size_t)(row0 + i * 4) * LATD + (kc + KCHUNK) + kk0);
        }

#pragma unroll
        for (int k4 = 0; k4 < KCHUNK / 4; ++k4) {
            // A (lat, 16x4): lane -> M=ll, K = kc + k4*4 + 2*lg + {0,1}
            const v2f a = *(const v2f*)&lds_lat[ll * LATD + kc + k4 * 4 + lg * 2];
            // B (W tile, 4x16): lane -> N=ll, K = k4*4 + 2*lg + {0,1}
            const v2f b = *(const v2f*)&lds_w[wave][buf][ll * KCHUNK + k4 * 4 + lg * 2];
            acc = __builtin_amdgcn_wmma_f32_16x16x4_f32(
                false, a, false, b, (short)0, acc, false, false);
        }
        buf ^= 1;
    }

    // Store: ks[m][col], m = r + 8*lg, col = c0 + ll (16-lane contiguous rows)
#pragma unroll
    for (int r = 0; r < 8; ++r)
        ks[(size_t)(r + 8 * lg) * NCOL + c0 + ll] = acc[r];
}

// ---------------------------------------------------------------------------
// Stage 2: per-sample 3x3 conv as implicit GEMM, compute-bound -> WMMA f32.
// Block: 1 sample, 16 fouts, 8 rows x 16 cols of output (8 waves, one output
// row-segment each). K-dim = FIN*9 = 1152, in fin chunks of CF=16 (144
// k-values -> 36 WMMAs per chunk, 288 per tile). Weight tiles staged with
// async b32 copies; x tile staged normally (needs zero-halo writes).
// ---------------------------------------------------------------------------
__global__ __launch_bounds__(256)
void dyn_conv_kernel(const float* __restrict__ x,
                     const float* __restrict__ ks,
                     float* __restrict__ y)
{
    __shared__ float lds_x[CF * 10 * 18];   // fin-chunk x (8+2 halo) x (16+2 halo)
    __shared__ float lds_w[16 * CF * 9];    // 16 fout x (CF*9) k-values

    const int tid  = threadIdx.x;
    const int wave = tid >> 5;
    const int lane = tid & 31;
    const int lg   = lane >> 4;
    const int ll   = lane & 15;

    const int sb  = blockIdx.x;             // 32 spatial tiles
    const int r0  = (sb >> 2) * 8;          // row base (8 row-groups)
    const int c0  = (sb & 3) * 16;          // col base (4 col-groups)
    const int fo0 = blockIdx.y * 16;        // fout tile
    const int b   = blockIdx.z;             // sample

    const float* ksb = ks + (size_t)b * NCOL;

    // C init = per-sample generated bias[fout]; M = fout = r + 8*lg.
    v8f acc;
#pragma unroll
    for (int r = 0; r < 8; ++r)
        acc[r] = ksb[KSZ + fo0 + r + 8 * lg];

    const int xbase = wave * 18 + ll;       // per-lane base into lds_x rows

    for (int fc = 0; fc < FIN; fc += CF) {
        __syncthreads();   // previous iteration's LDS reads done

        // Stage x tile (CF x 10 x 18) with zero halo (normal path: conditional).
        for (int i = tid; i < CF * 10 * 18; i += 256) {
            const int fin = i / 180;
            const int rem = i % 180;
            const int row = rem / 18;
            const int col = rem % 18;
            const int gr  = r0 + row - 1;
            const int gc  = c0 + col - 1;
            float v = 0.0f;
            if (gr >= 0 && gr < HH && gc >= 0 && gc < WW)
                v = x[(((size_t)b * FIN + fc + fin) * HH + gr) * WW + gc];
            lds_x[i] = v;
        }
        // Stage generated weights with async LDS copies (contiguous per fout).
        for (int i = tid; i < 16 * CF * 9; i += 256) {
            const int fo  = i / (CF * 9);
            const int rem = i % (CF * 9);
            async_copy_b32(lds_addr_of(&lds_w[i]),
                           ksb + ((size_t)(fo0 + fo) * FIN + fc) * 9 + rem);
        }
        wait_async0();
        __syncthreads();

        // Incremental (fin,tap) tracking: k advances by 4 per WMMA, tap wraps at 9.
        int t0 = lg * 2;                    // 0 or 2
        int f0 = 0;
#pragma unroll 4
        for (int k4 = 0; k4 < (CF * 9) / 4; ++k4) {   // 36
            // A (weights 16x4): lane -> M=ll (fout), K = k4*4 + 2*lg + {0,1}
            const v2f a = *(const v2f*)&lds_w[ll * (CF * 9) + k4 * 4 + lg * 2];
            // B (im2col 4x16): lane -> N=ll (pixel)
            const int ky0 = (t0 * 11) >> 5;           // t/3 for t in [0,8]
            const int kx0 = t0 - 3 * ky0;
            int t1 = t0 + 1, f1 = f0;
            if (t1 == 9) { t1 = 0; ++f1; }
            const int ky1 = (t1 * 11) >> 5;
            const int kx1 = t1 - 3 * ky1;
            v2f bm;
            bm.x = lds_x[f0 * 180 + ky0 * 18 + kx0 + xbase];
            bm.y = lds_x[f1 * 180 + ky1 * 18 + kx1 + xbase];
            acc = __builtin_amdgcn_wmma_f32_16x16x4_f32(
                false, a, false, bm, (short)0, acc, false, false);
            t0 += 4;
            if (t0 >= 9) { t0 -= 9; ++f0; }
        }
    }

    // Store: y[b][fout][r0+wave][c0+ll], fout = fo0 + r + 8*lg.
    const int row = r0 + wave;
#pragma unroll
    for (int r = 0; r < 8; ++r) {
        const int fo = fo0 + r + 8 * lg;
        y[(((size_t)b * FOUT + fo) * HH + row) * WW + c0 + ll] = acc[r];
    }
}

extern "C" void kernel_launch(void* const* d_in, const int* in_sizes, int n_in,
                              void* d_out, int out_size, void* d_ws, size_t ws_size,
                              hipStream_t stream)
{
    (void)in_sizes; (void)n_in; (void)out_size; (void)ws_size;
    const float* x    = (const float*)d_in[0];   // (16,128,64,64)
    const float* lat  = (const float*)d_in[1];   // (16,512)
    const float* Wm   = (const float*)d_in[2];   // (147584,512)
    const float* bias = (const float*)d_in[3];   // (147584,)
    float* y  = (float*)d_out;                   // (16,128,64,64)
    float* ks = (float*)d_ws;                    // (16,147584) scratch, 9.0 MB

    // Stage 1: 9224 column-tiles / 8 waves = 1153 blocks.
    hyper_gemm_kernel<<<dim3(NCOL / 128), 256, 0, stream>>>(lat, Wm, bias, ks);
    // Stage 2: 32 spatial tiles x 8 fout tiles x 16 samples.
    dyn_conv_kernel<<<dim3(32, 8, BATCH), 256, 0, stream>>>(x, ks, y);
}